// VQEmbeddingEMA_22806276342344
// MI455X (gfx1250) — compile-verified
//
#include <hip/hip_runtime.h>
#include <hip/hip_bf16.h>
#include <math.h>

typedef __attribute__((ext_vector_type(2))) float v2f;
typedef __attribute__((ext_vector_type(8))) float v8f;

#define D      1024
#define M      256
#define NT     25600            // 64*400
#define TOTAL  26214400ull      // NT * D
#define NBLK   1600             // NT / 16

// ---------------------------------------------------------------------------
// Kernel A: normalize codebook rows, compute ||en_m||^2, zero histogram.
// grid = 256 blocks (one per code), block = 256 threads (4 floats each).
// ---------------------------------------------------------------------------
__global__ __launch_bounds__(256)
void vq_normalize(const float* __restrict__ emb, float* __restrict__ en,
                  float* __restrict__ en_sq, unsigned* __restrict__ hist) {
    const int m   = blockIdx.x;
    const int tid = threadIdx.x;
    const int lane = tid & 31, w = tid >> 5;

    const float4 v = ((const float4*)(emb + (size_t)m * D))[tid];
    float s = v.x * v.x + v.y * v.y + v.z * v.z + v.w * v.w;

    __shared__ float red[8];
    __shared__ float inv_sh;
    #pragma unroll
    for (int off = 16; off >= 1; off >>= 1) s += __shfl_down(s, off, 32);
    if (lane == 0) red[w] = s;
    __syncthreads();
    if (tid == 0) {
        float t = 0.f;
        #pragma unroll
        for (int i = 0; i < 8; ++i) t += red[i];
        const float inv = 1.0f / (sqrtf(t) + 1e-4f);
        inv_sh   = inv;
        en_sq[m] = t * inv * inv;
        hist[m]  = 0u;
    }
    __syncthreads();
    const float inv = inv_sh;
    float4 o;
    o.x = v.x * inv; o.y = v.y * inv; o.z = v.z * inv; o.w = v.w * inv;
    ((float4*)(en + (size_t)m * D))[tid] = o;
}

// ---------------------------------------------------------------------------
// Kernel B: per 16-row tile of x: scores s[i][m] = ||en_m||^2 - 2 x_i.en_m via
// V_WMMA_F32_16X16X4_F32, argmin over m, gather + straight-through output,
// per-block loss partial, histogram update.
// grid = 1600 blocks, block = 512 threads = 16 waves (one wave per 16 columns).
// ---------------------------------------------------------------------------
__global__ __launch_bounds__(512)
void vq_assign(const float* __restrict__ x, const float* __restrict__ en,
               const float* __restrict__ en_sq, const float* __restrict__ emb,
               float* __restrict__ out, unsigned* __restrict__ hist,
               float* __restrict__ partial) {
    __shared__ float s_scores[16][M];   // 16 KB
    __shared__ int   s_idx[16];
    __shared__ float s_red[16];

    const int tid  = threadIdx.x;
    const int lane = tid & 31;
    const int w    = tid >> 5;          // wave id 0..15 -> column block
    const int colbase = w * 16;
    const int rc    = lane & 15;        // A: row index M; B: col index N
    const int half2 = (lane >> 4) << 1; // K sub-pair select: 0 or 2

    // A fragment: x[tile_row = rc][k0 + half2 + {0,1}]
    // B fragment: en[colbase + rc][k0 + half2 + {0,1}]  (B[k][n] = en[n][k])
    const size_t xoff = (size_t)(blockIdx.x * 16 + rc) * D + half2;
    const size_t eoff = (size_t)(colbase + rc) * D + half2;

    v8f acc = {0.f, 0.f, 0.f, 0.f, 0.f, 0.f, 0.f, 0.f};
    #pragma unroll 8
    for (int k0 = 0; k0 < D; k0 += 4) {
        v2f a = *(const v2f*)(x  + xoff + k0);
        v2f b = *(const v2f*)(en + eoff + k0);
        acc = __builtin_amdgcn_wmma_f32_16x16x4_f32(
            false, a, false, b, (short)0, acc, false, false);
    }

    // Scatter scores to LDS: C vgpr r holds (M = r + 8*(lane>=16), N = lane&15)
    const float esq = en_sq[colbase + rc];
    const int rb = (lane < 16) ? 0 : 8;
    #pragma unroll
    for (int r = 0; r < 8; ++r)
        s_scores[rb + r][colbase + rc] = esq - 2.0f * acc[r];
    __syncthreads();

    // Argmin: wave w reduces row w (first-occurrence tiebreak like jnp.argmin)
    {
        float best = s_scores[w][lane];
        int   bidx = lane;
        #pragma unroll
        for (int c = lane + 32; c < M; c += 32) {
            float v = s_scores[w][c];
            if (v < best) { best = v; bidx = c; }
        }
        #pragma unroll
        for (int off = 16; off >= 1; off >>= 1) {
            float ov = __shfl_down(best, off, 32);
            int   oi = __shfl_down(bidx, off, 32);
            if (ov < best || (ov == best && oi < bidx)) { best = ov; bidx = oi; }
        }
        if (lane == 0) {
            s_idx[w] = bidx;
            atomicAdd(&hist[bidx], 1u);
        }
    }
    __syncthreads();

    // Gather from UNnormalized codebook, mirror straight-through numerics,
    // accumulate commitment-loss partial. float4 vectorized: 4096 float4s.
    float lsum = 0.f;
    const float4* x4 = (const float4*)x;
    const float4* e4 = (const float4*)emb;
    float4*       o4 = (float4*)out;
    #pragma unroll
    for (int t = tid; t < 16 * (D / 4); t += 512) {
        const int row = t >> 8;          // 256 float4 per row
        const int d4  = t & 255;
        const size_t g = (size_t)(blockIdx.x * 16 + row) * (D / 4) + d4;
        const float4 xv = x4[g];
        const float4 q  = e4[(size_t)s_idx[row] * (D / 4) + d4];
        float4 o;
        o.x = 0.5f * ((xv.x + (q.x - xv.x)) + q.x);
        o.y = 0.5f * ((xv.y + (q.y - xv.y)) + q.y);
        o.z = 0.5f * ((xv.z + (q.z - xv.z)) + q.z);
        o.w = 0.5f * ((xv.w + (q.w - xv.w)) + q.w);
        o4[g] = o;
        const float dx = xv.x - q.x, dy = xv.y - q.y,
                    dz = xv.z - q.z, dw = xv.w - q.w;
        lsum += dx * dx + dy * dy + dz * dz + dw * dw;
    }
    #pragma unroll
    for (int off = 16; off >= 1; off >>= 1) lsum += __shfl_down(lsum, off, 32);
    if (lane == 0) s_red[w] = lsum;
    __syncthreads();
    if (tid == 0) {
        float t = 0.f;
        #pragma unroll
        for (int i = 0; i < 16; ++i) t += s_red[i];
        partial[blockIdx.x] = t;   // deterministic: private slot, summed later
    }
}

// ---------------------------------------------------------------------------
// Kernel C: loss = sum(partials)/TOTAL ; perplexity = exp(-sum p log(p+1e-10))
// 1 block x 256 threads.
// ---------------------------------------------------------------------------
__global__ __launch_bounds__(256)
void vq_finalize(const unsigned* __restrict__ hist,
                 const float* __restrict__ partial,
                 float* __restrict__ scal) {
    const int tid = threadIdx.x;
    const int lane = tid & 31, w = tid >> 5;

    const float p = (float)hist[tid] * (1.0f / (float)NT);
    float h = -p * logf(p + 1e-10f);
    float l = 0.f;
    for (int i = tid; i < NBLK; i += 256) l += partial[i];

    __shared__ float rh[8], rl[8];
    #pragma unroll
    for (int off = 16; off >= 1; off >>= 1) {
        h += __shfl_down(h, off, 32);
        l += __shfl_down(l, off, 32);
    }
    if (lane == 0) { rh[w] = h; rl[w] = l; }
    __syncthreads();
    if (tid == 0) {
        float hs = 0.f, ls = 0.f;
        #pragma unroll
        for (int i = 0; i < 8; ++i) { hs += rh[i]; ls += rl[i]; }
        scal[0] = ls / (float)TOTAL;  // commitment_loss
        scal[1] = expf(hs);           // perplexity
    }
}

// ---------------------------------------------------------------------------
extern "C" void kernel_launch(void* const* d_in, const int* in_sizes, int n_in,
                              void* d_out, int out_size, void* d_ws, size_t ws_size,
                              hipStream_t stream) {
    const float* x   = (const float*)d_in[0];   // (25600, 1024)
    const float* emb = (const float*)d_in[1];   // (256, 1024)
    float* out = (float*)d_out;                 // 26214400 + 2 scalars

    // workspace layout (floats)
    float*    ws      = (float*)d_ws;
    float*    en      = ws;                     // 256*1024
    float*    en_sq   = ws + (size_t)M * D;     // 256
    unsigned* hist    = (unsigned*)(en_sq + M); // 256
    float*    partial = (float*)(hist + M);     // 1600

    vq_normalize<<<M, 256, 0, stream>>>(emb, en, en_sq, hist);
    vq_assign<<<NBLK, 512, 0, stream>>>(x, en, en_sq, emb, out, hist, partial);
    vq_finalize<<<1, 256, 0, stream>>>(hist, partial, out + TOTAL);
}